// PointPillarScatterMAE_32066225832083
// MI455X (gfx1250) — compile-verified
//
#include <hip/hip_runtime.h>
#include <hip/hip_bf16.h>
#include <stdint.h>

// Problem constants (from the reference)
#define NXc   512
#define NYc   512
#define Cc    64
#define Bc    4
#define GRIDc (NXc * NYc)                      // 262144 cells per frame (nz==1)

#define SPATIAL_ELEMS ((size_t)Bc * Cc * GRIDc)   // 67,108,864
#define MAP_ELEMS     ((size_t)Bc * GRIDc)        //  1,048,576
#define OCC_OFF  (SPATIAL_ELEMS)
#define DEN_OFF  (OCC_OFF + MAP_ELEMS)
#define PTS_OFF  (DEN_OFF + MAP_ELEMS)
#define TOTAL_ELEMS (PTS_OFF + MAP_ELEMS)         // 70,254,592 floats
#define TOTAL_BYTES (TOTAL_ELEMS * 4ull)          // 281,018,368 bytes

#define CHUNK_BYTES 65536u                        // 64 KB LDS staging tile per WG
#define FILL_BLOCKS ((unsigned)(TOTAL_BYTES / CHUNK_BYTES))  // 4288, exact
static_assert(TOTAL_BYTES % CHUNK_BYTES == 0, "chunking must be exact");

typedef unsigned int v4u __attribute__((ext_vector_type(4)));
typedef int          v8i __attribute__((ext_vector_type(8)));
typedef int          v4i __attribute__((ext_vector_type(4)));

// ---------------------------------------------------------------------------
// Kernel 1: zero-fill the entire 281 MB output via the Tensor Data Mover.
// Each workgroup zeroes a 64 KB LDS buffer, then one wave issues a single
// TENSOR_STORE_FROM_LDS covering its 64 KB global chunk. DMA engine streams
// the stores; wave just waits on TENSORcnt.
// ---------------------------------------------------------------------------
__global__ __launch_bounds__(256) void tdm_zero_fill(float* __restrict__ out)
{
    __shared__ float lds_zero[CHUNK_BYTES / 4];   // 64 KB of LDS

    // Zero the LDS staging tile (256 threads x 16 float4 = 64 KB).
    float4* l4 = (float4*)lds_zero;
    #pragma unroll
    for (int i = 0; i < (int)((CHUNK_BYTES / 16) / 256); ++i) {
        float4 z = {0.0f, 0.0f, 0.0f, 0.0f};
        l4[threadIdx.x + i * 256] = z;
    }
    __syncthreads();

    if (threadIdx.x == 0) {
        uint64_t gaddr   = (uint64_t)(uintptr_t)out + (uint64_t)blockIdx.x * CHUNK_BYTES;
        uint32_t lds_off = (uint32_t)(uintptr_t)&lds_zero[0];   // flat addr[31:0] == LDS byte offset

        // ---- D# group 0 (128b): count=1 | lds_addr | global_addr | type=2 ----
        v4u g0;
        g0.x = 1u;                                           // count=1 (valid), user mode
        g0.y = lds_off;                                      // lds_addr (bytes)
        g0.z = (uint32_t)(gaddr & 0xFFFFFFFFu);              // global_addr[31:0]
        g0.w = (uint32_t)((gaddr >> 32) & 0x01FFFFFFu)       // global_addr[56:32]
             | (2u << 30);                                   // type = 2 ("image")

        // ---- D# group 1 (256b): 2-D tile, data_size=8B, tile==tensor ----
        // tensor_dim0 = tile_dim0 = 8192 (x 8B = 64 KB row), dim1 = 1,
        // strides = 8192 (unused with 1 row, but kept consistent).
        v8i g1;
        g1[0] = (int)(3u << 16);        // wg_mask=0 | data_size=3 (8B) | no flags
        g1[1] = (int)(8192u << 16);     // atomic_barrier_addr=0 | tensor_dim0[15:0]=8192
        g1[2] = (int)(1u << 16);        // tensor_dim0[31:16]=0  | tensor_dim1[15:0]=1
        g1[3] = (int)(8192u << 16);     // tensor_dim1[31:16]=0  | tile_dim0=8192
        g1[4] = 1;                      // tile_dim1=1 | tile_dim2=0
        g1[5] = 8192;                   // tensor_dim0_stride[31:0]
        g1[6] = (int)(8192u << 16);     // stride0[47:32]=0 | tensor_dim1_stride[15:0]
        g1[7] = 0;                      // tensor_dim1_stride[47:16]=0

        v4i g2 = {0, 0, 0, 0};          // 2-D: groups 2/3 unused
        v4i g3 = {0, 0, 0, 0};
        v8i gx = {0, 0, 0, 0, 0, 0, 0, 0};  // 6-arg toolchain: extra group, unused

        __builtin_amdgcn_tensor_store_from_lds(g0, g1, g2, g3, gx, 0);
        __builtin_amdgcn_s_wait_tensorcnt(0);
    }
}

// ---------------------------------------------------------------------------
// Kernel 2: per-pillar BEV maps (occupancy / density / points_number).
// One thread per pillar; int4 coord load, three 4B scatter stores.
// ---------------------------------------------------------------------------
__global__ __launch_bounds__(256) void scatter_maps(const int* __restrict__ coords,
                                                    const int* __restrict__ npts,
                                                    float* __restrict__ out, int M)
{
    int m = blockIdx.x * 256 + threadIdx.x;
    if (m >= M) return;
    int4 c = ((const int4*)coords)[m];          // (b, z, y, x)
    size_t flat = (size_t)c.x * GRIDc + c.y + (size_t)c.z * NXc + c.w;
    int n = npts[m];
    out[OCC_OFF + flat] = 1.0f;
    out[DEN_OFF + flat] = (float)n * (1.0f / 32.0f);
    out[PTS_OFF + flat] = (float)n;             // int map emitted as float
}

// ---------------------------------------------------------------------------
// Kernel 3: feature scatter for selected pillars.
// 16 threads per selected pillar: coalesced float4 read of the 256 B feature
// row; each thread stores its 4 channels at GRIDc (1 MB) stride into the
// channel-major spatial tensor [B, C, NY*NX].
// ---------------------------------------------------------------------------
__global__ __launch_bounds__(256) void scatter_feats(const float* __restrict__ feats,
                                                     const int* __restrict__ coords,
                                                     const int* __restrict__ sel,
                                                     float* __restrict__ out, int K)
{
    int t = blockIdx.x * 256 + threadIdx.x;
    int k = t >> 4;                             // pillar index into select_index
    int q = t & 15;                             // which float4 of the 64-ch row
    if (k >= K) return;
    int s = sel[k];
    int4 c = ((const int4*)coords)[s];          // (b, z, y, x)
    int p  = c.y + c.z * NXc + c.w;             // within-frame cell
    float4 f = ((const float4*)feats)[(size_t)s * 16 + q];
    float* base = out + ((size_t)(c.x * Cc + q * 4)) * GRIDc + p;
    base[0 * (size_t)GRIDc] = f.x;
    base[1 * (size_t)GRIDc] = f.y;
    base[2 * (size_t)GRIDc] = f.z;
    base[3 * (size_t)GRIDc] = f.w;
}

extern "C" void kernel_launch(void* const* d_in, const int* in_sizes, int n_in,
                              void* d_out, int out_size, void* d_ws, size_t ws_size,
                              hipStream_t stream)
{
    const float* feats  = (const float*)d_in[0];   // [M, 64] f32
    const int*   coords = (const int*)d_in[1];     // [M, 4]  i32
    const int*   npts   = (const int*)d_in[2];     // [M]     i32
    const int*   sel    = (const int*)d_in[3];     // [K]     i32
    float*       out    = (float*)d_out;

    const int M = in_sizes[2];
    const int K = in_sizes[3];

    // 1) DMA zero-fill of the whole output (TDM, 64 KB per workgroup).
    hipLaunchKernelGGL(tdm_zero_fill, dim3(FILL_BLOCKS), dim3(256), 0, stream, out);
    // 2) occupancy / density / points maps.
    hipLaunchKernelGGL(scatter_maps, dim3((M + 255) / 256), dim3(256), 0, stream,
                       coords, npts, out, M);
    // 3) selected pillar feature scatter.
    hipLaunchKernelGGL(scatter_feats, dim3((K * 16 + 255) / 256), dim3(256), 0, stream,
                       feats, coords, sel, out, K);
}